// Pool_5557687681651
// MI455X (gfx1250) — compile-verified
//
#include <hip/hip_runtime.h>
#include <hip/hip_bf16.h>

// Problem constants from the reference
#define BB    4
#define NN    4096
#define DD    256
#define SS    64
#define KK    1024

typedef __attribute__((ext_vector_type(2))) float v2f;
typedef __attribute__((ext_vector_type(8))) float v8f;

// ---------------------------------------------------------------------------
// Kernel 1: scores[b,n] = sigmoid( max_s  h[b,n,:] . sf[b,s,:] )
// One wave32 per 16-row tile of n.  Uses V_WMMA_F32_16X16X4_F32.
//
// A (16x4 f32) lane layout: lane L -> row M = L%16, holds K = 2*(L/16) and +1.
// B (4x16 f32) lane layout: lane L -> col N = L%16, holds K = 2*(L/16) and +1.
// C/D (16x16 f32): VGPR r, lanes 0-15 -> M=r, lanes 16-31 -> M=8+r; N = L%16.
// ---------------------------------------------------------------------------
__global__ __launch_bounds__(32)
void score_kernel(const float* __restrict__ h,
                  const float* __restrict__ sf,
                  float* __restrict__ scores) {
    __shared__ float4 sh4[16 * DD / 4];   // 16 rows of h       (16 KB)
    __shared__ float4 ssf4[16 * DD / 4];  // 16 rows of sf tile (16 KB)
    float* sh  = (float*)sh4;
    float* ssf = (float*)ssf4;

    const int lane = threadIdx.x;            // 0..31
    const int blk  = blockIdx.x;             // B * N/16 blocks
    const int b    = blk >> 8;               // N/16 == 256 tiles per batch
    const int n0   = (blk & 255) * 16;

    // Stage 16 contiguous h rows (coalesced float4)
    const float4* hsrc = (const float4*)(h + ((size_t)b * NN + n0) * DD);
    for (int t = lane; t < 16 * DD / 4; t += 32) sh4[t] = hsrc[t];

    const int m  = lane & 15;          // row (for A) / col (for B) inside tile
    const int kb = (lane >> 4) << 1;   // K sub-offset: 0 or 2

    v8f m8;
    for (int r = 0; r < 8; ++r) m8[r] = -3.0e38f;

    for (int s0 = 0; s0 < SS / 16; ++s0) {
        __syncthreads();
        const float4* ssrc = (const float4*)(sf + ((size_t)b * SS + s0 * 16) * DD);
        for (int t = lane; t < 16 * DD / 4; t += 32) ssf4[t] = ssrc[t];
        __syncthreads();

        v8f c = {};
        for (int k = 0; k < DD; k += 4) {
            v2f a, bv;
            a.x  = sh[m * DD + k + kb];
            a.y  = sh[m * DD + k + kb + 1];
            bv.x = ssf[m * DD + k + kb];
            bv.y = ssf[m * DD + k + kb + 1];
            // D = A(16x4) * B(4x16) + C, exact f32 accumulation
            c = __builtin_amdgcn_wmma_f32_16x16x4_f32(
                    /*neg_a=*/false, a, /*neg_b=*/false, bv,
                    /*c_mod=*/(short)0, c, /*reuse_a=*/false, /*reuse_b=*/false);
        }
        for (int r = 0; r < 8; ++r) m8[r] = fmaxf(m8[r], c[r]);
    }

    // Row-max across the 16 columns: reduce over lanes within each 16-lane half
    for (int mask = 8; mask >= 1; mask >>= 1)
        for (int r = 0; r < 8; ++r)
            m8[r] = fmaxf(m8[r], __shfl_xor(m8[r], mask, 32));

    if ((lane & 15) == 0) {
        const int mb = (lane >> 4) * 8;   // lanes 0..15 -> rows 0..7, 16..31 -> 8..15
        for (int r = 0; r < 8; ++r) {
            float w = m8[r];
            scores[(size_t)b * NN + n0 + mb + r] = 1.0f / (1.0f + __expf(-w));
        }
    }
}

// ---------------------------------------------------------------------------
// Kernel 2: per-batch top-K via full bitonic sort of 4096 64-bit keys in LDS.
// key = score_bits << 32 | (0xFFFFFFFF - idx)   (score>0 so float bits are
// monotone); descending sort == score desc, index asc (jax top_k tie rule).
// One 1024-thread workgroup per batch.
// ---------------------------------------------------------------------------
__global__ __launch_bounds__(1024)
void topk_kernel(const float* __restrict__ scores,
                 int* __restrict__ idx_out,
                 float* __restrict__ val_out) {
    __shared__ unsigned long long keys[NN];   // 32 KB
    const int b   = blockIdx.x;
    const int tid = threadIdx.x;
    const float* sc = scores + (size_t)b * NN;

    for (int i = tid; i < NN; i += 1024) {
        unsigned int fb = __float_as_uint(sc[i]);
        keys[i] = ((unsigned long long)fb << 32) |
                  (unsigned long long)(0xFFFFFFFFu - (unsigned int)i);
    }
    __syncthreads();

    for (int size = 2; size <= NN; size <<= 1) {
        for (int stride = size >> 1; stride > 0; stride >>= 1) {
            for (int t = tid; t < NN / 2; t += 1024) {
                int i = 2 * stride * (t / stride) + (t % stride);
                int j = i + stride;
                unsigned long long a = keys[i];
                unsigned long long c = keys[j];
                bool descBlock = ((i & size) == 0);
                bool doSwap    = descBlock ? (a < c) : (a > c);
                if (doSwap) { keys[i] = c; keys[j] = a; }
            }
            __syncthreads();
        }
    }

    if (tid < KK) {
        unsigned long long k = keys[tid];
        idx_out[b * KK + tid] = (int)(0xFFFFFFFFu - (unsigned int)k);
        val_out[b * KK + tid] = __uint_as_float((unsigned int)(k >> 32));
    }
}

// ---------------------------------------------------------------------------
// Kernel 3: bandwidth phase.
//   new_h[b,i,:]      = h[b, idx_i, :] * val_i
//   g_section[b,i,j]  = g1[b, idx_i, idx_j]
//   g_sentence[b,i,j] = g2[b, idx_i, idx_j]
// One 256-thread block per (b,i); index vector cached in LDS; writes fully
// coalesced, reads served from L2 (gathered rows fit in 192 MB L2).
// ---------------------------------------------------------------------------
__global__ __launch_bounds__(256)
void gather_kernel(const float* __restrict__ g1,
                   const float* __restrict__ g2,
                   const float* __restrict__ h,
                   const int* __restrict__ idx,
                   const float* __restrict__ vals,
                   float* __restrict__ out_gsec,
                   float* __restrict__ out_gsen,
                   float* __restrict__ out_newh) {
    __shared__ int sidx[KK];
    const int blk = blockIdx.x;
    const int b   = blk >> 10;        // / KK
    const int i   = blk & (KK - 1);
    const int tid = threadIdx.x;      // 0..255

    for (int t = tid; t < KK; t += 256) sidx[t] = idx[b * KK + t];
    __syncthreads();

    const int   ii = sidx[i];
    const float vi = vals[b * KK + i];

    const float* r1 = g1 + ((size_t)b * NN + ii) * NN;
    const float* r2 = g2 + ((size_t)b * NN + ii) * NN;
    __builtin_prefetch(r1 + tid * 16, 0, 0);   // global_prefetch_b8
    __builtin_prefetch(r2 + tid * 16, 0, 0);

    // new_h: D == blockDim, one element per thread, coalesced
    const float* hrow = h + ((size_t)b * NN + ii) * DD;
    out_newh[((size_t)b * KK + i) * DD + tid] = hrow[tid] * vi;

    float* o1 = out_gsec + ((size_t)b * KK + i) * KK;
    float* o2 = out_gsen + ((size_t)b * KK + i) * KK;
    for (int j = tid; j < KK; j += 256) {
        int jj = sidx[j];
        o1[j] = r1[jj];
        o2[j] = r2[jj];
    }
}

// ---------------------------------------------------------------------------
extern "C" void kernel_launch(void* const* d_in, const int* in_sizes, int n_in,
                              void* d_out, int out_size, void* d_ws, size_t ws_size,
                              hipStream_t stream) {
    const float* g1 = (const float*)d_in[0];
    const float* g2 = (const float*)d_in[1];
    const float* h  = (const float*)d_in[2];
    const float* sf = (const float*)d_in[3];

    // Outputs concatenated flat in return order: g_section, g_sentence, new_h
    float* out_gsec = (float*)d_out;
    float* out_gsen = out_gsec + (size_t)BB * KK * KK;
    float* out_newh = out_gsen + (size_t)BB * KK * KK;

    // Workspace: scores (B*N f32), idx (B*K i32), vals (B*K f32)  ~96 KB
    float* ws_scores = (float*)d_ws;
    int*   ws_idx    = (int*)(ws_scores + (size_t)BB * NN);
    float* ws_vals   = (float*)(ws_idx + (size_t)BB * KK);

    score_kernel<<<dim3(BB * (NN / 16)), dim3(32), 0, stream>>>(h, sf, ws_scores);
    topk_kernel<<<dim3(BB), dim3(1024), 0, stream>>>(ws_scores, ws_idx, ws_vals);
    gather_kernel<<<dim3(BB * KK), dim3(256), 0, stream>>>(
        g1, g2, h, ws_idx, ws_vals, out_gsec, out_gsen, out_newh);
}